// ElementLoss_46720654246270
// MI455X (gfx1250) — compile-verified
//
#include <hip/hip_runtime.h>
#include <math.h>

// ---------------- problem constants (fixed by reference) ----------------
#define T_DIM 4096
#define D_DIM 256
#define R_DIM 4
#define TILE  16
#define NWAVES 8                 // 256 threads = 8 wave32
#define NJB   (T_DIM / TILE)     // 256 column blocks

typedef __attribute__((ext_vector_type(2))) float v2f;
typedef __attribute__((ext_vector_type(8))) float v8f;

// f32 WMMA: D(16x16) = A(16x4) * B(4x16) + C ; A,B = 2 VGPRs/lane, C/D = 8
__device__ inline v8f wmma_f32x4(v2f a, v2f b, v8f c) {
    return __builtin_amdgcn_wmma_f32_16x16x4_f32(
        /*neg_a=*/false, a, /*neg_b=*/false, b,
        /*c_mod=*/(short)0, c, /*reuse_a=*/false, /*reuse_b=*/false);
}

// ---------------- kernel 0: zero the scalar output ----------------
__global__ void zero_out_kernel(float* out) {
    if (threadIdx.x == 0) out[0] = 0.0f;
}

// ---------------- kernel 1: per-row sums of M (4096 rows) ----------------
__global__ void rowsum_kernel(const float* __restrict__ M, float* __restrict__ rs) {
    int row  = blockIdx.x * 8 + (threadIdx.x >> 5);   // 8 rows per 256-thread block
    int lane = threadIdx.x & 31;
    const float* p = M + (size_t)row * D_DIM;
    float s = 0.0f;
    for (int c = lane; c < D_DIM; c += 32) s += p[c];
    #pragma unroll
    for (int off = 16; off > 0; off >>= 1) s += __shfl_xor(s, off, 32);
    if (lane == 0) rs[row] = s;
}

// ---------------- kernel 2: fused pairwise-std GEMM + top3 + loss ----------------
__global__ __launch_bounds__(256) void
pairstd_loss_kernel(const float* __restrict__ X,
                    const float* __restrict__ A,
                    const float* __restrict__ M,
                    const int*   __restrict__ pos,     // row_elements_pos (T x R)
                    const float* __restrict__ rowsum,  // per-row sum(M)
                    float*       __restrict__ out) {
    // LDS: phase 1 = I-panel {M, Xm, X2m} (3*4096 floats = 48 KB)
    //      phase 2 = aliased top3 dump (6144 vals + 6144 cols = 48 KB)
    __shared__ float smem[3 * TILE * D_DIM];          // 12288 floats
    float* sM   = smem;
    float* sXm  = smem + TILE * D_DIM;
    float* sX2m = smem + 2 * TILE * D_DIM;

    const int tid  = threadIdx.x;
    const int i0   = blockIdx.x * TILE;

    // ---- stage I-panel (coalesced) ----
    #pragma unroll
    for (int e = 0; e < TILE; ++e) {
        int idx = tid + 256 * e;                      // 0 .. 4095
        float x = X[(size_t)i0 * D_DIM + idx];
        float m = M[(size_t)i0 * D_DIM + idx];
        float xm = x * m;
        sM[idx]   = m;
        sXm[idx]  = xm;
        sX2m[idx] = x * xm;
    }
    __syncthreads();

    const int wave = tid >> 5;
    const int lane = tid & 31;
    const int half = lane >> 4;                       // 0 or 1
    const int ln   = lane & 15;                       // 0..15
    const int colk = 2 * half;                        // K sub-offset (0 or 2)

    // lane-local running top-3 (smallest scores) for rows r+8*half
    float best[8][3];
    int   bcol[8][3];
    #pragma unroll
    for (int r = 0; r < 8; ++r)
        #pragma unroll
        for (int s = 0; s < 3; ++s) { best[r][s] = __builtin_inff(); bcol[r][s] = -1; }

    float rs_i[8];
    #pragma unroll
    for (int r = 0; r < 8; ++r) rs_i[r] = rowsum[i0 + r + 8 * half];

    for (int jb = wave; jb < NJB; jb += NWAVES) {
        const int j0 = jb * TILE;
        const int jg = j0 + ln;                       // this lane's column index
        const float* Xj = X + (size_t)jg * D_DIM;
        const float* Mj = M + (size_t)jg * D_DIM;

        v8f acc_n  = {}; v8f acc_p  = {}; v8f acc_pt = {};
        v8f acc_q  = {}; v8f acc_r1 = {}; v8f acc_r2 = {};

        for (int k0 = 0; k0 < D_DIM; k0 += 4) {
            const int c = k0 + colk;
            v2f am   = *(const v2f*)(sM   + ln * D_DIM + c);
            v2f axm  = *(const v2f*)(sXm  + ln * D_DIM + c);
            v2f ax2m = *(const v2f*)(sX2m + ln * D_DIM + c);
            v2f bx   = *(const v2f*)(Xj + c);
            v2f bm   = *(const v2f*)(Mj + c);
            v2f bxm  = bx * bm;
            v2f bx2m = bx * bxm;

            acc_n  = wmma_f32x4(am,   bm,   acc_n);   // M  . M^T
            acc_p  = wmma_f32x4(axm,  bm,   acc_p);   // Xm . M^T
            acc_pt = wmma_f32x4(am,   bxm,  acc_pt);  // M  . Xm^T
            acc_q  = wmma_f32x4(axm,  bxm,  acc_q);   // Xm . Xm^T
            acc_r1 = wmma_f32x4(ax2m, bm,   acc_r1);  // X2m. M^T
            acc_r2 = wmma_f32x4(am,   bx2m, acc_r2);  // M  . X2m^T
        }

        const float rsj = rowsum[jg];
        #pragma unroll
        for (int r = 0; r < 8; ++r) {
            const int ig = i0 + r + 8 * half;
            float n  = acc_n[r];
            float s1 = acc_p[r] - acc_pt[r];
            float s2 = acc_r1[r] - 2.0f * acc_q[r] + acc_r2[r];
            float nsafe = fmaxf(n, 2.0f);
            float var = (s2 - s1 * s1 / nsafe) / (nsafe - 1.0f);
            float sd  = sqrtf(fmaxf(var, 1e-12f));
            float neq = rs_i[r] + rsj - 2.0f * n;
            float score = (neq > 0.5f && ig != jg) ? sd : __builtin_inff();
            // top-3 insertion (ascending)
            if (score < best[r][2]) {
                if (score < best[r][1]) {
                    best[r][2] = best[r][1]; bcol[r][2] = bcol[r][1];
                    if (score < best[r][0]) {
                        best[r][1] = best[r][0]; bcol[r][1] = bcol[r][0];
                        best[r][0] = score;      bcol[r][0] = jg;
                    } else { best[r][1] = score; bcol[r][1] = jg; }
                } else { best[r][2] = score; bcol[r][2] = jg; }
            }
        }
    }

    // ---- dump lane-local top3s into LDS (alias panel region) ----
    __syncthreads();                                  // panel reads done
    float* tv = smem;                                 // 6144 floats
    int*   tc = (int*)(smem + 6144);                  // 6144 ints
    const int base = ((wave * 32 + lane) * 8) * 3;
    #pragma unroll
    for (int r = 0; r < 8; ++r)
        #pragma unroll
        for (int s = 0; s < 3; ++s) {
            tv[base + r * 3 + s] = best[r][s];
            tc[base + r * 3 + s] = bcol[r][s];
        }
    __syncthreads();

    // ---- 16 threads: merge 8 waves x 16 lanes x 3 slots per row, then loss ----
    if (tid < TILE) {
        const int i = tid;
        const int r = i & 7;
        const int h = i >> 3;
        float b0 = __builtin_inff(), b1 = b0, b2 = b0;
        int   c0 = -1, c1 = -1, c2 = -1;
        for (int w = 0; w < NWAVES; ++w) {
            for (int l = 16 * h; l < 16 * h + 16; ++l) {
                const int eb = ((w * 32 + l) * 8 + r) * 3;
                #pragma unroll
                for (int s = 0; s < 3; ++s) {
                    float v = tv[eb + s];
                    int   c = tc[eb + s];
                    if (v < b2) {
                        if (v < b1) {
                            b2 = b1; c2 = c1;
                            if (v < b0) { b1 = b0; c1 = c0; b0 = v; c0 = c; }
                            else        { b1 = v;  c1 = c; }
                        } else { b2 = v; c2 = c; }
                    }
                }
            }
        }
        const int ig = i0 + i;
        if (c0 >= 0 && b0 < __builtin_inff()) {
            // softmax over (-b0, -b1, -b2), max = -b0
            float e0 = 1.0f;
            float e1 = (c1 >= 0) ? __expf(b0 - b1) : 0.0f;
            float e2 = (c2 >= 0) ? __expf(b0 - b2) : 0.0f;
            float denom = e0 + e1 + e2;

            float Ar[R_DIM];
            int   pr[R_DIM];
            #pragma unroll
            for (int k = 0; k < R_DIM; ++k) {
                Ar[k] = A[ig * R_DIM + k];
                pr[k] = pos[ig * R_DIM + k];
            }
            float loss = 0.0f;
            const int   cands[3] = { c0, c1, c2 };
            const float ws[3]    = { e0 / denom, e1 / denom, e2 / denom };
            #pragma unroll
            for (int t = 0; t < 3; ++t) {
                if (cands[t] < 0 || ws[t] <= 0.0f) continue;
                const float* xr = X + (size_t)cands[t] * D_DIM;
                float acc = 0.0f;
                #pragma unroll
                for (int k = 0; k < R_DIM; ++k)
                    acc += fabsf(Ar[k] - xr[pr[k]]);
                loss += ws[t] * acc;
            }
            atomicAdd(out, loss);
        }
    }
}

// ---------------- host-side launch ----------------
extern "C" void kernel_launch(void* const* d_in, const int* in_sizes, int n_in,
                              void* d_out, int out_size, void* d_ws, size_t ws_size,
                              hipStream_t stream) {
    const float* X   = (const float*)d_in[0];   // (4096, 256)
    const float* A   = (const float*)d_in[1];   // (16384,)
    const float* M   = (const float*)d_in[2];   // (4096, 256)
    const int*   pos = (const int*)  d_in[5];   // (4096, 4)
    float* out = (float*)d_out;                 // scalar
    float* rs  = (float*)d_ws;                  // 4096 floats of scratch

    zero_out_kernel<<<1, 32, 0, stream>>>(out);
    rowsum_kernel<<<T_DIM / 8, 256, 0, stream>>>(M, rs);
    pairstd_loss_kernel<<<T_DIM / TILE, 256, 0, stream>>>(X, A, M, pos, rs, out);
}